// Method_50818053046387
// MI455X (gfx1250) — compile-verified
//
#include <hip/hip_runtime.h>

#define IC   16
#define BS   16
#define BB   8
#define HH   96
#define WW2  96
#define HW   (HH*WW2)        // 9216
#define CH   (IC*BS)         // 256
#define CSTR (BS*HW)         // 147456, stride of IC-channel within batch
#define SPB  (CH*HW)         // 2359296 elems per batch
#define TOT  (BB*SPB)        // 18874368
#define EPSV 1e-5f

typedef __attribute__((ext_vector_type(2))) float v2f;
typedef __attribute__((ext_vector_type(8))) float v8f;

__device__ __forceinline__ float gelu_exact(float x) {
    return 0.5f * x * (1.0f + erff(x * 0.70710678118654752440f));
}

// ---------------- stats helpers ----------------

__global__ void k_zero_stats(float* p) {
    if (threadIdx.x < 512) p[threadIdx.x] = 0.f;
}

__device__ __forceinline__ void block_reduce_stats(float v, int ch,
                                                   float* gsum, float* gsq) {
    __shared__ float s1[256], s2[256];
    int tid = threadIdx.x;
    s1[tid] = v; s2[tid] = v * v;
    __syncthreads();
    for (int o = 128; o > 0; o >>= 1) {
        if (tid < o) { s1[tid] += s1[tid + o]; s2[tid] += s2[tid + o]; }
        __syncthreads();
    }
    if (tid == 0) { atomicAdd(&gsum[ch], s1[0]); atomicAdd(&gsq[ch], s2[0]); }
}

// finalize per-IC (16 ch) stats, replicate affine across BS -> 256 slots
__global__ void k_fin_ic(const float* gsum, const float* gsq,
                         const float* gamma, const float* beta,
                         float* sc, float* sh, float count) {
    int tid = threadIdx.x;
    if (tid < 16) {
        float m = gsum[tid] / count;
        float v = gsq[tid] / count - m * m;
        float s = gamma[tid] * rsqrtf(v + EPSV);
        float h = beta[tid] - m * s;
        for (int r = 0; r < 16; r++) { sc[tid * 16 + r] = s; sh[tid * 16 + r] = h; }
    }
}

__global__ void k_fin_256(const float* gsum, const float* gsq,
                          const float* gamma, const float* beta,
                          float* sc, float* sh, float count) {
    int tid = threadIdx.x;
    float m = gsum[tid] / count;
    float v = gsq[tid] / count - m * m;
    float s = gamma[tid] * rsqrtf(v + EPSV);
    sc[tid] = s;
    sh[tid] = beta[tid] - m * s;
}

// ---------------- stage 0: depthwise 3x3x3 conv + GELU, emit stats ----------------
// grid (36, BS, B*IC), block 256
__global__ void k_conv3d(const float* __restrict__ x, const float* __restrict__ w3,
                         const float* __restrict__ b3, float* __restrict__ out,
                         float* gsum, float* gsq) {
    int tid = threadIdx.x;
    int pos = blockIdx.x * 256 + tid;
    int h = pos / WW2, w = pos - h * WW2;
    int bs = blockIdx.y;
    int b = blockIdx.z >> 4, ic = blockIdx.z & 15;
    const float* xb = x + (size_t)(b * IC + ic) * CSTR;
    const float* wk = w3 + ic * 27;
    float acc = 0.f;
    for (int kd = 0; kd < 3; kd++) {
        int d2 = bs + kd - 1;
        if ((unsigned)d2 >= (unsigned)BS) continue;
        const float* xd = xb + d2 * HW;
        for (int kh = 0; kh < 3; kh++) {
            int h2 = h + kh - 1; if ((unsigned)h2 >= (unsigned)HH) continue;
            for (int kw = 0; kw < 3; kw++) {
                int w2 = w + kw - 1; if ((unsigned)w2 >= (unsigned)WW2) continue;
                acc += wk[kd * 9 + kh * 3 + kw] * xd[h2 * WW2 + w2];
            }
        }
    }
    float v = gelu_exact(acc + b3[ic]);
    out[(size_t)(b * IC + ic) * CSTR + bs * HW + pos] = v;
    block_reduce_stats(v, ic, gsum, gsq);
}

// ---------------- bandwise: depthwise 3x3 conv on BN'd input + GELU ----------------
// grid (36, 256, B), block 256
__global__ void k_bandwise(const float* __restrict__ in, const float* __restrict__ sc,
                           const float* __restrict__ sh, const float* __restrict__ wk9,
                           const float* __restrict__ bias, float* __restrict__ out,
                           float* gsum, float* gsq) {
    int tid = threadIdx.x;
    int pos = blockIdx.x * 256 + tid;
    int h = pos / WW2, w = pos - h * WW2;
    int ch = blockIdx.y, b = blockIdx.z;
    const float* ip = in + (size_t)(b * CH + ch) * HW;
    float s = sc[ch], t = sh[ch];
    const float* wk = wk9 + ch * 9;
    float acc = 0.f;
    for (int kh = 0; kh < 3; kh++) {
        int h2 = h + kh - 1; if ((unsigned)h2 >= (unsigned)HH) continue;
        for (int kw = 0; kw < 3; kw++) {
            int w2 = w + kw - 1; if ((unsigned)w2 >= (unsigned)WW2) continue;
            acc += wk[kh * 3 + kw] * (ip[h2 * WW2 + w2] * s + t);
        }
    }
    float v = gelu_exact(acc + bias[ch]);
    out[(size_t)(b * CH + ch) * HW + pos] = v;
    block_reduce_stats(v, ch, gsum, gsq);
}

// ---------------- pixelwise: per-block 16x16 GEMM via WMMA f32 16x16x4, in place ----
// grid (8, IC, B), block 256 (8 waves, 9 tiles of 16 positions each)
__global__ void k_pixelwise(float* io, const float* __restrict__ sc,
                            const float* __restrict__ sh,
                            const float* __restrict__ pw, const float* __restrict__ pb,
                            float* gsum, float* gsq) {
    __shared__ float lsS[16], lsQ[16];
    int tid = threadIdx.x;
    if (tid < 16) { lsS[tid] = 0.f; lsQ[tid] = 0.f; }
    __syncthreads();
    int lane = tid & 31, wave = tid >> 5;
    int g = blockIdx.y, b = blockIdx.z;
    int lm = lane & 15, kh2 = lane >> 4;
    // A = weights, M(=o)xK(=i): lane holds M=lm, k = 4s + 2*kh2 + j
    float aw[8], bc[8];
#pragma unroll
    for (int s = 0; s < 4; s++) {
        aw[2 * s]     = pw[(g * 16 + lm) * 16 + 4 * s + 2 * kh2];
        aw[2 * s + 1] = pw[(g * 16 + lm) * 16 + 4 * s + 2 * kh2 + 1];
    }
#pragma unroll
    for (int j = 0; j < 8; j++) bc[j] = pb[g * 16 + (j + 8 * kh2)];
    float stS[8], stQ[8];
#pragma unroll
    for (int j = 0; j < 8; j++) { stS[j] = 0.f; stQ[j] = 0.f; }
    float* base = io + (size_t)(b * IC + g) * CSTR;  // channel i at base + i*HW
    int t0 = (blockIdx.x * 8 + wave) * 9;
    for (int t = 0; t < 9; t++) {
        int qb = (t0 + t) * 16;
        v8f c = {};
#pragma unroll
        for (int s = 0; s < 4; s++) {
            int k0 = 4 * s + 2 * kh2;
            float b0 = base[k0 * HW + qb + lm] * sc[g * 16 + k0] + sh[g * 16 + k0];
            float b1 = base[(k0 + 1) * HW + qb + lm] * sc[g * 16 + k0 + 1] + sh[g * 16 + k0 + 1];
            v2f A; A.x = aw[2 * s]; A.y = aw[2 * s + 1];
            v2f Bv; Bv.x = b0; Bv.y = b1;
            c = __builtin_amdgcn_wmma_f32_16x16x4_f32(false, A, false, Bv,
                                                      (short)0, c, false, false);
        }
#pragma unroll
        for (int j = 0; j < 8; j++) {
            int Mc = j + 8 * kh2;
            float v = gelu_exact(c[j] + bc[j]);
            base[Mc * HW + qb + lm] = v;
            stS[j] += v; stQ[j] += v * v;
        }
    }
#pragma unroll
    for (int j = 0; j < 8; j++)
        for (int m = 1; m < 16; m <<= 1) {
            stS[j] += __shfl_xor(stS[j], m, 32);
            stQ[j] += __shfl_xor(stQ[j], m, 32);
        }
    if (lm == 0) {
#pragma unroll
        for (int j = 0; j < 8; j++) {
            int Mc = j + 8 * kh2;
            atomicAdd(&lsS[Mc], stS[j]); atomicAdd(&lsQ[Mc], stQ[j]);
        }
    }
    __syncthreads();
    if (tid < 16) { atomicAdd(&gsum[g * 16 + tid], lsS[tid]); atomicAdd(&gsq[g * 16 + tid], lsQ[tid]); }
}

// ---------------- pointwise: residual + 16x16 GEMM over IC via WMMA, in place into Q --
// grid (128, B), block 256
__global__ void k_pointwise(const float* __restrict__ inP, const float* __restrict__ scP,
                            const float* __restrict__ shP,
                            float* ioQ, const float* __restrict__ scQ,
                            const float* __restrict__ shQ,
                            const float* __restrict__ pw, const float* __restrict__ pb,
                            float* gsum, float* gsq) {
    __shared__ float lsS[16], lsQ2[16];
    int tid = threadIdx.x;
    if (tid < 16) { lsS[tid] = 0.f; lsQ2[tid] = 0.f; }
    __syncthreads();
    int lane = tid & 31, wave = tid >> 5;
    int b = blockIdx.y;
    int lm = lane & 15, kh2 = lane >> 4;
    float aw[8], bc[8];
#pragma unroll
    for (int s = 0; s < 4; s++) {
        aw[2 * s]     = pw[lm * 16 + 4 * s + 2 * kh2];      // pt_w[o=lm][c=k]
        aw[2 * s + 1] = pw[lm * 16 + 4 * s + 2 * kh2 + 1];
    }
#pragma unroll
    for (int j = 0; j < 8; j++) bc[j] = pb[j + 8 * kh2];
    float stS[8], stQ[8];
#pragma unroll
    for (int j = 0; j < 8; j++) { stS[j] = 0.f; stQ[j] = 0.f; }
    const float* Pb = inP + (size_t)b * SPB;
    float* Qb = ioQ + (size_t)b * SPB;
    int t0 = (blockIdx.x * 8 + wave) * 9;
    for (int t = 0; t < 9; t++) {
        int qb = (t0 + t) * 16;
        int bs = qb / HW;                // constant within a 16-wide tile
        v8f c = {};
#pragma unroll
        for (int s = 0; s < 4; s++) {
            int k0 = 4 * s + 2 * kh2;
            int a0 = k0 * CSTR + qb + lm;
            int a1 = a0 + CSTR;
            int c0 = k0 * 16 + bs, c1 = c0 + 16;
            float v0 = Pb[a0] * scP[c0] + shP[c0] + Qb[a0] * scQ[c0] + shQ[c0];
            float v1 = Pb[a1] * scP[c1] + shP[c1] + Qb[a1] * scQ[c1] + shQ[c1];
            v2f A; A.x = aw[2 * s]; A.y = aw[2 * s + 1];
            v2f Bv; Bv.x = v0; Bv.y = v1;
            c = __builtin_amdgcn_wmma_f32_16x16x4_f32(false, A, false, Bv,
                                                      (short)0, c, false, false);
        }
#pragma unroll
        for (int j = 0; j < 8; j++) {
            int Mc = j + 8 * kh2;
            float v = gelu_exact(c[j] + bc[j]);
            Qb[Mc * CSTR + qb + lm] = v;
            stS[j] += v; stQ[j] += v * v;
        }
    }
#pragma unroll
    for (int j = 0; j < 8; j++)
        for (int m = 1; m < 16; m <<= 1) {
            stS[j] += __shfl_xor(stS[j], m, 32);
            stQ[j] += __shfl_xor(stQ[j], m, 32);
        }
    if (lm == 0) {
#pragma unroll
        for (int j = 0; j < 8; j++) {
            int Mc = j + 8 * kh2;
            atomicAdd(&lsS[Mc], stS[j]); atomicAdd(&lsQ2[Mc], stQ[j]);
        }
    }
    __syncthreads();
    if (tid < 16) { atomicAdd(&gsum[tid], lsS[tid]); atomicAdd(&gsq[tid], lsQ2[tid]); }
}

// ---------------- final BN apply ----------------
__global__ void k_apply(const float* __restrict__ in, const float* __restrict__ sc,
                        const float* __restrict__ sh, float* __restrict__ out) {
    int i = blockIdx.x * 256 + threadIdx.x;
    int ch = (i / HW) & 255;
    out[i] = in[i] * sc[ch] + sh[ch];
}

extern "C" void kernel_launch(void* const* d_in, const int* in_sizes, int n_in,
                              void* d_out, int out_size, void* d_ws, size_t ws_size,
                              hipStream_t stream) {
    (void)in_sizes; (void)n_in; (void)out_size; (void)ws_size;
    const float* x     = (const float*)d_in[0];
    const float* w3    = (const float*)d_in[1];
    const float* b3    = (const float*)d_in[2];
    const float* g3    = (const float*)d_in[3];
    const float* be3   = (const float*)d_in[4];
    const float* bw_w  = (const float*)d_in[5];
    const float* bw_b  = (const float*)d_in[6];
    const float* bw_g  = (const float*)d_in[7];
    const float* bw_be = (const float*)d_in[8];
    const float* px_w  = (const float*)d_in[9];
    const float* px_b  = (const float*)d_in[10];
    const float* px_g  = (const float*)d_in[11];
    const float* px_be = (const float*)d_in[12];
    const float* pt_w  = (const float*)d_in[13];
    const float* pt_b  = (const float*)d_in[14];
    const float* pt_g  = (const float*)d_in[15];
    const float* pt_be = (const float*)d_in[16];

    float* W0    = (float*)d_ws;
    float* stats = W0 + TOT;
    float* gsum  = stats;
    float* gsq   = stats + 256;
    float* aA_s  = stats + 512,  *aA_h = stats + 768;   // affine of current "P"
    float* aB_s  = stats + 1024, *aB_h = stats + 1280;  // bandwise affine
    float* aC_s  = stats + 1536, *aC_h = stats + 1792;  // pixelwise affine
    float* W1    = (float*)d_out;

    const float cnt_ic  = (float)(BB * BS * HW);  // 1179648
    const float cnt_256 = (float)(BB * HW);       // 73728

    k_zero_stats<<<1, 512, 0, stream>>>(stats);
    k_conv3d<<<dim3(36, 16, BB * IC), 256, 0, stream>>>(x, w3, b3, W0, gsum, gsq);
    k_fin_ic<<<1, 64, 0, stream>>>(gsum, gsq, g3, be3, aA_s, aA_h, cnt_ic);

    float* P = W0;
    float* Q = W1;
    for (int d = 0; d < 4; d++) {
        k_zero_stats<<<1, 512, 0, stream>>>(stats);
        k_bandwise<<<dim3(36, 256, BB), 256, 0, stream>>>(
            P, aA_s, aA_h, bw_w + d * CH * 9, bw_b + d * CH, Q, gsum, gsq);
        k_fin_256<<<1, 256, 0, stream>>>(gsum, gsq, bw_g + d * CH, bw_be + d * CH,
                                         aB_s, aB_h, cnt_256);

        k_zero_stats<<<1, 512, 0, stream>>>(stats);
        k_pixelwise<<<dim3(8, IC, BB), 256, 0, stream>>>(
            Q, aB_s, aB_h, px_w + d * IC * BS * BS, px_b + d * CH, gsum, gsq);
        k_fin_256<<<1, 256, 0, stream>>>(gsum, gsq, px_g + d * CH, px_be + d * CH,
                                         aC_s, aC_h, cnt_256);

        k_zero_stats<<<1, 512, 0, stream>>>(stats);
        k_pointwise<<<dim3(128, BB), 256, 0, stream>>>(
            P, aA_s, aA_h, Q, aC_s, aC_h, pt_w + d * IC * IC, pt_b + d * IC, gsum, gsq);
        k_fin_ic<<<1, 64, 0, stream>>>(gsum, gsq, pt_g + d * IC, pt_be + d * IC,
                                       aA_s, aA_h, cnt_ic);

        float* tmp = P; P = Q; Q = tmp;
    }
    // after 4 swaps P == W0 (ws); write normalized result to d_out
    k_apply<<<dim3(TOT / 256), 256, 0, stream>>>(P, aA_s, aA_h, (float*)d_out);
}